// MSDeformableAttn_23184233463853
// MI455X (gfx1250) — compile-verified
//
#include <hip/hip_runtime.h>
#include <math.h>

typedef __attribute__((ext_vector_type(2))) float v2f;
typedef __attribute__((ext_vector_type(8))) float v8f;

#define NB      2
#define QLEN    8192
#define NQ      (NB * QLEN)        // 16384
#define EMB     256
#define HEADS   8
#define LEVELS  4
#define POINTS  4
#define HD      32
#define SPATIAL 5440               // 4096 + 1024 + 256 + 64
#define OA_COLS 384                // 256 offset + 128 attn logits

// ---------------------------------------------------------------------------
// One wave computes one 16x16 fp32 output tile with V_WMMA_F32_16X16X4_F32.
// A element (m,k) at A[m*a_row_stride + k*a_k_stride]  (handles both row-major
// query rows and channel-major feature maps).  B row-major (k*b_k_stride+col).
// A frag layout (ISA 7.12.2): lanes 0-15 -> K=0,1 ; lanes 16-31 -> K=2,3.
// ---------------------------------------------------------------------------
__device__ __forceinline__ v8f wmma_f32_tile(
    const float* __restrict__ A, int a_row_stride, int a_k_stride, int m0,
    const float* __restrict__ B, int b_k_stride, int col0,
    int K, int lane)
{
  const int half = lane >> 4;
  const int lm   = lane & 15;
  const int koff = half << 1;
  const float* __restrict__ arow = A + (m0 + lm) * a_row_stride;
  const float* __restrict__ bcol = B + col0 + lm;
  v8f c = {};
  #pragma unroll 4
  for (int kk = 0; kk < K; kk += 4) {
    const int k0 = kk + koff;
    v2f a, b;
    a[0] = arow[(k0    ) * a_k_stride];
    a[1] = arow[(k0 + 1) * a_k_stride];
    b[0] = bcol[(k0    ) * b_k_stride];
    b[1] = bcol[(k0 + 1) * b_k_stride];
    c = __builtin_amdgcn_wmma_f32_16x16x4_f32(false, a, false, b,
                                              (short)0, c, false, false);
  }
  return c;
}

// ---------------------------------------------------------------------------
// Kernel 1: offset + attention-logit projection.
//   oa[row, 0:256]   = query @ W_off  + b_off
//   oa[row, 256:384] = query @ W_attn + b_attn
// waves = 1024 row-tiles * 24 col-tiles = 24576
// ---------------------------------------------------------------------------
__global__ void msda_oa_proj_kernel(const float* __restrict__ query,
                                    const float* __restrict__ Woff,
                                    const float* __restrict__ boff,
                                    const float* __restrict__ Wattn,
                                    const float* __restrict__ battn,
                                    float* __restrict__ oa)
{
  const int wave = (blockIdx.x * blockDim.x + threadIdx.x) >> 5;
  const int lane = threadIdx.x & 31;
  const int col_tile = wave % 24;
  const int m_tile   = wave / 24;
  const int col0 = col_tile * 16;

  const float* B; int ldb, colB; const float* bias;
  if (col0 < 256) { B = Woff;  ldb = 256; colB = col0;       bias = boff  + col0;       }
  else            { B = Wattn; ldb = 128; colB = col0 - 256; bias = battn + (col0-256); }

  v8f c = wmma_f32_tile(query, EMB, 1, m_tile * 16, B, ldb, colB, EMB, lane);

  const int half = lane >> 4, lm = lane & 15;
  const float bb = bias[lm];
  #pragma unroll
  for (int r = 0; r < 8; ++r) {
    const int row = m_tile * 16 + r + half * 8;
    oa[row * OA_COLS + col0 + lm] = c[r] + bb;
  }
}

// ---------------------------------------------------------------------------
// Kernel 2: value projection (1x1 conv) per level.
//   valbuf[(n*SPATIAL + lvl_off + p)*256 + d] = sum_c feat[n,c,p]*Wv[c,d] + bv[d]
// Per batch: 256+64+16+4 = 340 row-tiles, 16 col-tiles -> 10880 waves total.
// ---------------------------------------------------------------------------
__global__ void msda_val_proj_kernel(const float* __restrict__ f0,
                                     const float* __restrict__ f1,
                                     const float* __restrict__ f2,
                                     const float* __restrict__ f3,
                                     const float* __restrict__ Wv,
                                     const float* __restrict__ bv,
                                     float* __restrict__ valbuf)
{
  const int wave = (blockIdx.x * blockDim.x + threadIdx.x) >> 5;
  const int lane = threadIdx.x & 31;
  const int col_tile = wave & 15;
  const int rest = wave >> 4;           // n*340 + t
  const int n = rest / 340;
  const int t = rest - n * 340;

  const float* feat; int HW, lvl_off, mt;
  if      (t < 256) { feat = f0; HW = 4096; lvl_off = 0;    mt = t;       }
  else if (t < 320) { feat = f1; HW = 1024; lvl_off = 4096; mt = t - 256; }
  else if (t < 336) { feat = f2; HW = 256;  lvl_off = 5120; mt = t - 320; }
  else              { feat = f3; HW = 64;   lvl_off = 5376; mt = t - 336; }
  feat += n * EMB * HW;                 // (C, HW) channel-major for batch n

  const int m0   = mt * 16;
  const int col0 = col_tile * 16;
  // A(m,k) = feat[k*HW + m]  -> row stride 1, k stride HW
  v8f c = wmma_f32_tile(feat, 1, HW, m0, Wv, EMB, col0, EMB, lane);

  const int half = lane >> 4, lm = lane & 15;
  const int d = col0 + lm;
  const float bb = bv[d];
  #pragma unroll
  for (int r = 0; r < 8; ++r) {
    const int p = m0 + r + half * 8;
    valbuf[(n * SPATIAL + lvl_off + p) * EMB + d] = c[r] + bb;
  }
}

// ---------------------------------------------------------------------------
// Kernel 3: softmax + bilinear sampling + weighted sum.
// One wave32 per (n, q, head); one lane per channel hd (HD == 32).
// Each corner gather = one coalesced 128B transaction (channels contiguous).
// ---------------------------------------------------------------------------
__global__ void msda_sample_kernel(const float* __restrict__ oa,    // (NQ,384)
                                   const float* __restrict__ refp,  // (NQ,2)
                                   const float* __restrict__ valbuf,// (N,SPATIAL,256)
                                   float* __restrict__ attn_out)    // (NQ,256)
{
  const int gw   = (blockIdx.x * blockDim.x + threadIdx.x) >> 5;
  const int lane = threadIdx.x & 31;
  const int h = gw & 7;
  const int q = (gw >> 3) & (QLEN - 1);
  const int n = gw >> 16;               // gw / (HEADS*QLEN)
  const int row = n * QLEN + q;
  const float* __restrict__ oarow = oa + (size_t)row * OA_COLS;

  // softmax over the 16 (level,point) logits of this head
  float logit[16];
  float mx = -3.402823466e38f;
  #pragma unroll
  for (int i = 0; i < 16; ++i) {
    logit[i] = oarow[256 + h * 16 + i];
    mx = fmaxf(mx, logit[i]);
  }
  float ssum = 0.f;
  #pragma unroll
  for (int i = 0; i < 16; ++i) { logit[i] = __expf(logit[i] - mx); ssum += logit[i]; }
  const float inv = 1.0f / ssum;        // ssum >= 1 (max logit contributes 1)

  const float rx = refp[row * 2 + 0];
  const float ry = refp[row * 2 + 1];

  float acc = 0.f;
  int lvl_off = 0;
  #pragma unroll
  for (int l = 0; l < LEVELS; ++l) {
    const int Wl = 64 >> l, Hl = 64 >> l;
    const float* __restrict__ vbase =
        valbuf + ((size_t)(n * SPATIAL + lvl_off)) * EMB + h * HD + lane;
    #pragma unroll
    for (int p = 0; p < POINTS; ++p) {
      const int oidx = (h * LEVELS + l) * POINTS + p;
      const float ox = oarow[oidx * 2 + 0];
      const float oy = oarow[oidx * 2 + 1];
      // loc in [0,1] -> grid in [-1,1] -> pixel coords (align_corners=False)
      const float x = (rx + ox) * (float)Wl - 0.5f;
      const float y = (ry + oy) * (float)Hl - 0.5f;
      const float x0f = floorf(x), y0f = floorf(y);
      const float wx = x - x0f,   wy = y - y0f;
      const int x0 = (int)x0f, y0 = (int)y0f;
      const int x1 = x0 + 1,   y1 = y0 + 1;
      const float vx0 = (x0 >= 0 && x0 < Wl) ? 1.f : 0.f;
      const float vx1 = (x1 >= 0 && x1 < Wl) ? 1.f : 0.f;
      const float vy0 = (y0 >= 0 && y0 < Hl) ? 1.f : 0.f;
      const float vy1 = (y1 >= 0 && y1 < Hl) ? 1.f : 0.f;
      const int cx0 = min(max(x0, 0), Wl - 1);
      const int cx1 = min(max(x1, 0), Wl - 1);
      const int cy0 = min(max(y0, 0), Hl - 1);
      const int cy1 = min(max(y1, 0), Hl - 1);
      const float g00 = vbase[(cy0 * Wl + cx0) * EMB];
      const float g10 = vbase[(cy0 * Wl + cx1) * EMB];
      const float g01 = vbase[(cy1 * Wl + cx0) * EMB];
      const float g11 = vbase[(cy1 * Wl + cx1) * EMB];
      const float bil = g00 * ((1.f - wx) * (1.f - wy) * vx0 * vy0)
                      + g10 * ((      wx) * (1.f - wy) * vx1 * vy0)
                      + g01 * ((1.f - wx) * (      wy) * vx0 * vy1)
                      + g11 * ((      wx) * (      wy) * vx1 * vy1);
      acc = fmaf(logit[l * POINTS + p] * inv, bil, acc);
    }
    lvl_off += Wl * Hl;
  }
  attn_out[(size_t)row * EMB + h * HD + lane] = acc;
}

// ---------------------------------------------------------------------------
// Kernel 4: output projection  out = attn_out @ W_out + b_out
// waves = 1024 row-tiles * 16 col-tiles = 16384
// ---------------------------------------------------------------------------
__global__ void msda_out_proj_kernel(const float* __restrict__ attn_in,
                                     const float* __restrict__ Wout,
                                     const float* __restrict__ bout,
                                     float* __restrict__ out)
{
  const int wave = (blockIdx.x * blockDim.x + threadIdx.x) >> 5;
  const int lane = threadIdx.x & 31;
  const int col_tile = wave & 15;
  const int m_tile   = wave >> 4;
  const int col0 = col_tile * 16;

  v8f c = wmma_f32_tile(attn_in, EMB, 1, m_tile * 16, Wout, EMB, col0, EMB, lane);

  const int half = lane >> 4, lm = lane & 15;
  const float bb = bout[col0 + lm];
  #pragma unroll
  for (int r = 0; r < 8; ++r) {
    const int row = m_tile * 16 + r + half * 8;
    out[(size_t)row * EMB + col0 + lm] = c[r] + bb;
  }
}

// ---------------------------------------------------------------------------
extern "C" void kernel_launch(void* const* d_in, const int* in_sizes, int n_in,
                              void* d_out, int out_size, void* d_ws, size_t ws_size,
                              hipStream_t stream) {
  const float* query = (const float*)d_in[0];
  const float* f0    = (const float*)d_in[1];
  const float* f1    = (const float*)d_in[2];
  const float* f2    = (const float*)d_in[3];
  const float* f3    = (const float*)d_in[4];
  const float* refp  = (const float*)d_in[5];
  const float* Woff  = (const float*)d_in[6];
  const float* boff  = (const float*)d_in[7];
  const float* Wattn = (const float*)d_in[8];
  const float* battn = (const float*)d_in[9];
  const float* Wval  = (const float*)d_in[10];
  const float* bval  = (const float*)d_in[11];
  const float* Wout  = (const float*)d_in[12];
  const float* bout  = (const float*)d_in[13];

  float* ws     = (float*)d_ws;
  float* oa     = ws;                                   // NQ * 384
  float* valbuf = oa + (size_t)NQ * OA_COLS;            // NB * SPATIAL * 256
  float* attn   = valbuf + (size_t)NB * SPATIAL * EMB;  // NQ * 256

  // 1) offset/attn projection: 24576 waves, 8 waves/block
  msda_oa_proj_kernel<<<3072, 256, 0, stream>>>(query, Woff, boff, Wattn, battn, oa);
  // 2) value projection: 10880 waves, 4 waves/block
  msda_val_proj_kernel<<<2720, 128, 0, stream>>>(f0, f1, f2, f3, Wval, bval, valbuf);
  // 3) sampling: 131072 waves (one per n,q,head), 8 waves/block
  msda_sample_kernel<<<16384, 256, 0, stream>>>(oa, refp, valbuf, attn);
  // 4) output projection: 16384 waves
  msda_out_proj_kernel<<<2048, 256, 0, stream>>>(attn, Wout, bout, (float*)d_out);
}